// SymbolicTraversal_24507083391244
// MI455X (gfx1250) — compile-verified
//
#include <hip/hip_runtime.h>
#include <stdint.h>

// ---------------------------------------------------------------------------
// SymbolicTraversal: per-sample relation-masked scatter-max over an edge list.
//   out[b, dst[e]] = max(0, max_{e: type[e]==r_index[b]} h[b, src[e]])
// HBM-bound on streaming edge data (~38.4 MB/call @ 23.3 TB/s).
// Edge stream is staged LDS-double-buffered with CDNA5 async-to-LDS DMA.
// ---------------------------------------------------------------------------

#define BLOCK 256
#define EPT   4                  // edges per thread per tile
#define TILE  (BLOCK * EPT)      // 1024 edges per tile
#define NREL  64                 // relation vocabulary (r_index, edge_type < 64)

typedef int v4i __attribute__((vector_size(16)));           // b128 payload type
typedef __attribute__((address_space(1))) v4i glb_v4i;
typedef __attribute__((address_space(3))) v4i lds_v4i;

// generic->global: numerically identical 64-bit address.
__device__ __forceinline__ glb_v4i* as_glb(const void* p) {
    return (glb_v4i*)(uint64_t)p;
}
// generic->LDS: low 32 bits of the generic address are the wave-relative LDS
// byte offset (flat-aperture rule).
__device__ __forceinline__ lds_v4i* as_lds(const void* p) {
    return (lds_v4i*)(uint32_t)(uint64_t)p;
}

#if defined(__has_builtin) && __has_builtin(__builtin_amdgcn_global_load_async_to_lds_b128)
#define ASYNC_LOAD_B128(gp, lp) \
    __builtin_amdgcn_global_load_async_to_lds_b128((gp), (lp), 0, 0)
#else
#define ASYNC_LOAD_B128(gp, lp) \
    asm volatile("global_load_async_to_lds_b128 %0, %1, off" :: "v"(lp), "v"(gp) : "memory")
#endif

#if defined(__has_builtin) && __has_builtin(__builtin_amdgcn_s_wait_asynccnt)
#define WAIT_ASYNC(n) __builtin_amdgcn_s_wait_asynccnt(n)
#else
#define WAIT_ASYNC(n) asm volatile("s_wait_asynccnt %0" :: "n"(n) : "memory")
#endif

__global__ void zero_i32(int* __restrict__ out, int n) {
    int i = blockIdx.x * blockDim.x + threadIdx.x;
    if (i < n) out[i] = 0;
}

__global__ __launch_bounds__(BLOCK) void edge_relmax(
    const float* __restrict__ h,      // [B, N]
    const int*   __restrict__ esrc,   // [E]
    const int*   __restrict__ edst,   // [E]
    const int*   __restrict__ etyp,   // [E]
    const int*   __restrict__ rix,    // [B]
    int*         __restrict__ out,    // [B, N] as float bits (init 0)
    int E, int N, int Bn, int ntFull)
{
    __shared__ unsigned lut[NREL];
    __shared__ __align__(16) int sbuf[2][3][TILE];   // [buf][src/dst/typ][edge]

    // Build relation -> sample-bitmask LUT once per block.
    if (threadIdx.x < NREL) {
        unsigned m = 0;
        for (int b = 0; b < Bn; ++b)
            if (rix[b] == (int)threadIdx.x) m |= (1u << b);
        lut[threadIdx.x] = m;
    }
    __syncthreads();

    const int slot = threadIdx.x * EPT;

    auto issue = [&](long long tile, int buf) {
        long long e0 = tile * (long long)TILE + slot;
        ASYNC_LOAD_B128(as_glb(esrc + e0), as_lds(&sbuf[buf][0][slot]));
        ASYNC_LOAD_B128(as_glb(edst + e0), as_lds(&sbuf[buf][1][slot]));
        ASYNC_LOAD_B128(as_glb(etyp + e0), as_lds(&sbuf[buf][2][slot]));
    };

    auto process_edge = [&](int sv, int dv, int tv) {
        unsigned m = lut[tv & (NREL - 1)];
        while (m) {
            int b = __ffs(m) - 1;
            m &= m - 1;
            float v = h[(size_t)b * N + sv];
            // signed-int max on float bits: exact for init==0.0f (clamp>=0)
            atomicMax(out + (size_t)b * N + dv, __float_as_int(v));
        }
    };

    long long tile = blockIdx.x;
    if (tile < ntFull) {
        int cur = 0;
        issue(tile, cur);
        for (;;) {
            long long nxt = tile + (long long)gridDim.x;
            if (nxt < ntFull) {
                issue(nxt, cur ^ 1);   // keep next tile's DMA in flight
                WAIT_ASYNC(3);         // wait only for the current tile's 3 ops
            } else {
                WAIT_ASYNC(0);
            }
            // each thread consumes only its own LDS slots -> no barrier needed
            int4 s4 = *(const int4*)&sbuf[cur][0][slot];
            int4 d4 = *(const int4*)&sbuf[cur][1][slot];
            int4 t4 = *(const int4*)&sbuf[cur][2][slot];
            process_edge(s4.x, d4.x, t4.x);
            process_edge(s4.y, d4.y, t4.y);
            process_edge(s4.z, d4.z, t4.z);
            process_edge(s4.w, d4.w, t4.w);
            if (nxt >= ntFull) break;
            tile = nxt;
            cur ^= 1;
        }
    }

    // Remainder edges (E not a multiple of TILE): plain global loads.
    for (long long e = (long long)ntFull * TILE
                      + (long long)blockIdx.x * BLOCK + threadIdx.x;
         e < E;
         e += (long long)gridDim.x * BLOCK) {
        process_edge(esrc[e], edst[e], etyp[e]);
    }
}

extern "C" void kernel_launch(void* const* d_in, const int* in_sizes, int n_in,
                              void* d_out, int out_size, void* d_ws, size_t ws_size,
                              hipStream_t stream) {
    (void)n_in; (void)d_ws; (void)ws_size;
    const float* h    = (const float*)d_in[0];   // [B, N]
    const int*   eidx = (const int*)d_in[1];     // [2, E]
    const int*   etyp = (const int*)d_in[2];     // [E]
    const int*   rix  = (const int*)d_in[3];     // [B]
    int* out = (int*)d_out;                      // [B, N] float32 bits

    const int Bn = in_sizes[3];
    const int E  = in_sizes[2];
    const int N  = in_sizes[0] / (Bn > 0 ? Bn : 1);

    // 1) init output to 0.0f (subsumes clamp(min=0) under atomic-max)
    {
        int threads = 256;
        int blocks  = (out_size + threads - 1) / threads;
        zero_i32<<<blocks, threads, 0, stream>>>(out, out_size);
    }

    // 2) stream edges
    const int ntFull = E / TILE;
    int grid = 1024;                       // ~3 tiles/block -> DMA double-buffer pipelines
    if (ntFull > 0 && ntFull < grid) grid = ntFull;
    if (ntFull == 0) grid = (E + BLOCK - 1) / BLOCK > 0 ? (E + BLOCK - 1) / BLOCK : 1;

    edge_relmax<<<grid, BLOCK, 0, stream>>>(h, eidx, eidx + E, etyp, rix, out,
                                            E, N, Bn, ntFull);
}